// HTSATNet_86346022519278
// MI455X (gfx1250) — compile-verified
//
#include <hip/hip_runtime.h>
#include <math.h>

typedef __attribute__((ext_vector_type(2))) float v2f;
typedef __attribute__((ext_vector_type(8))) float v8f;

#define EPSV   1e-5f
#define Nn     32
#define CIN    64
#define Tt     64
#define Vv     25
#define TV     1600      // Tt*Vv
#define Ll     6
#define Ss     3
#define INTER  64
#define COUT   256
#define C4     64
#define NEGINF (-3.4e38f)

// --------------------------------------------------------------------------
// WMMA f32 GEMM with fused per-channel statistics.
//   Y[n,o,m] = sum_k W[o,k] * X[n,k,m] + bias[o]
//   X: [Nb, K, M] (k-stride = M), W: [O, K] row-major.
//   K and M are template constants so every k-step load/store offset is a
//   compile-time immediate (no per-iteration address arithmetic).
//   One wave = 16(O) x 64(M) macro-tile: per K-step 1 A-load + 8 B-loads +
//   4x V_WMMA_F32_16X16X4_F32 sharing the A operand.
//   Epilogue accumulates masked per-channel sum/sumsq via lane-shuffle
//   reduction + global f32 atomics into statsBuf[0..O) / statsBuf[O..2O).
// --------------------------------------------------------------------------
template<int K, int M>
__global__ __launch_bounds__(32)
void wmma_gemm_kernel(const float* __restrict__ X, const float* __restrict__ W,
                      const float* __restrict__ bias, float* __restrict__ Y,
                      int O, long xBatch, long yBatch, float* __restrict__ statsBuf)
{
    const int lane = threadIdx.x;       // 0..31
    const int col  = lane & 15;
    const int half = lane >> 4;         // 0/1
    const int mBase = blockIdx.x * 64;
    const int oBase = blockIdx.y * 16;
    const float* Xn = X + (long)blockIdx.z * xBatch;
    float*       Yn = Y + (long)blockIdx.z * yBatch;

    const int c0 = mBase + col;
    int bc0 = c0;      if (bc0 >= M) bc0 = M - 1;   // clamped B columns
    int bc1 = c0 + 16; if (bc1 >= M) bc1 = M - 1;
    int bc2 = c0 + 32; if (bc2 >= M) bc2 = M - 1;
    int bc3 = c0 + 48; if (bc3 >= M) bc3 = M - 1;

    const int arow = oBase + col;                        // A row (output channel)
    const float* wp  = W + (long)arow * K + 2 * half;    // lanes 0-15: k,k+1; 16-31: k+2,k+3
    const float* xr  = Xn + (long)(2 * half) * M;
    const float* xp0 = xr + bc0;
    const float* xp1 = xr + bc1;
    const float* xp2 = xr + bc2;
    const float* xp3 = xr + bc3;

    v8f acc0 = {}, acc1 = {}, acc2 = {}, acc3 = {};
    #pragma unroll 8
    for (int k0 = 0; k0 < K; k0 += 4) {
        v2f a, b0, b1, b2, b3;
        a.x  = wp[k0];           a.y  = wp[k0 + 1];
        b0.x = xp0[k0 * M];      b0.y = xp0[(k0 + 1) * M];
        b1.x = xp1[k0 * M];      b1.y = xp1[(k0 + 1) * M];
        b2.x = xp2[k0 * M];      b2.y = xp2[(k0 + 1) * M];
        b3.x = xp3[k0 * M];      b3.y = xp3[(k0 + 1) * M];
        acc0 = __builtin_amdgcn_wmma_f32_16x16x4_f32(false, a, false, b0, (short)0, acc0, false, false);
        acc1 = __builtin_amdgcn_wmma_f32_16x16x4_f32(false, a, false, b1, (short)0, acc1, false, false);
        acc2 = __builtin_amdgcn_wmma_f32_16x16x4_f32(false, a, false, b2, (short)0, acc2, false, false);
        acc3 = __builtin_amdgcn_wmma_f32_16x16x4_f32(false, a, false, b3, (short)0, acc3, false, false);
    }

    const bool s0 = (c0 < M), s1 = (c0 + 16 < M), s2 = (c0 + 32 < M), s3 = (c0 + 48 < M);
    float* yp = Yn + (long)(oBase + 8 * half) * M + c0;   // row r at yp + r*M

    #pragma unroll
    for (int r = 0; r < 8; ++r) {
        const int orow = oBase + r + 8 * half;
        const float bv = bias[orow];
        const float y0 = acc0[r] + bv, y1 = acc1[r] + bv;
        const float y2 = acc2[r] + bv, y3 = acc3[r] + bv;
        if (s0) yp[r * M]      = y0;
        if (s1) yp[r * M + 16] = y1;
        if (s2) yp[r * M + 32] = y2;
        if (s3) yp[r * M + 48] = y3;
        // masked partial stats for this output row
        float s = (s0 ? y0 : 0.f) + (s1 ? y1 : 0.f) + (s2 ? y2 : 0.f) + (s3 ? y3 : 0.f);
        float q = (s0 ? y0 * y0 : 0.f) + (s1 ? y1 * y1 : 0.f)
                + (s2 ? y2 * y2 : 0.f) + (s3 ? y3 * y3 : 0.f);
        #pragma unroll
        for (int mk = 8; mk >= 1; mk >>= 1) {
            s += __shfl_xor(s, mk, 16);
            q += __shfl_xor(q, mk, 16);
        }
        if (col == 0) {
            atomicAdd(&statsBuf[orow],     s);
            atomicAdd(&statsBuf[O + orow], q);
        }
    }
}

// statsBuf (sum, sumsq) -> mean, istd  (biased variance, + EPS, rsqrt)
__global__ void finalize_stats_kernel(const float* __restrict__ statsBuf,
                                      float* __restrict__ mean, float* __restrict__ istd,
                                      int O, float invTotal)
{
    int o = blockIdx.x * blockDim.x + threadIdx.x;
    if (o >= O) return;
    float m = statsBuf[o] * invTotal;
    float v = statsBuf[O + o] * invTotal - m * m;
    if (v < 0.f) v = 0.f;
    mean[o] = m;
    istd[o] = rsqrtf(v + EPSV);
}

// --------------------------------------------------------------------------
// Per-channel batch statistics over [Nb, O, M] (used for small edge tensors)
// --------------------------------------------------------------------------
__global__ __launch_bounds__(256)
void channel_stats_kernel(const float* __restrict__ X, float* __restrict__ mean,
                          float* __restrict__ istd, int Nb, int O, int M)
{
    const int o = blockIdx.x, tid = threadIdx.x;
    const long total = (long)Nb * M;
    float s = 0.f, q = 0.f;
    for (long i = tid; i < total; i += blockDim.x) {
        long n = i / M, m = i - n * M;
        float v = X[(n * O + o) * (long)M + m];
        s += v; q += v * v;
    }
    __shared__ float sh_s[256], sh_q[256];
    sh_s[tid] = s; sh_q[tid] = q;
    __syncthreads();
    for (int st = 128; st > 0; st >>= 1) {
        if (tid < st) { sh_s[tid] += sh_s[tid + st]; sh_q[tid] += sh_q[tid + st]; }
        __syncthreads();
    }
    if (tid == 0) {
        float m_ = sh_s[0] / (float)total;
        float v_ = sh_q[0] / (float)total - m_ * m_;
        if (v_ < 0.f) v_ = 0.f;
        mean[o] = m_;
        istd[o] = rsqrtf(v_ + EPSV);
    }
}

// BN (+optional ReLU) over contiguous [Nb, O, M]
__global__ void bn_act_kernel(const float* __restrict__ X, float* __restrict__ Y,
                              const float* __restrict__ mean, const float* __restrict__ istd,
                              const float* __restrict__ g, const float* __restrict__ b,
                              int O, long M, long total, int relu)
{
    long i = (long)blockIdx.x * blockDim.x + threadIdx.x;
    if (i >= total) return;
    int o = (int)((i / M) % O);
    float v = (X[i] - mean[o]) * istd[o] * g[o] + b[o];
    if (relu) v = fmaxf(v, 0.f);
    Y[i] = v;
}

// BN then scatter into outBig[N,COUT,L,T,V] at channel base c0, layer l
__global__ void bn_store_out_kernel(const float* __restrict__ X /*[N,INTER,TV]*/,
                                    float* __restrict__ outBig,
                                    const float* __restrict__ mean, const float* __restrict__ istd,
                                    const float* __restrict__ g, const float* __restrict__ b,
                                    int c0, int l)
{
    long i = (long)blockIdx.x * blockDim.x + threadIdx.x;
    if (i >= (long)Nn * INTER * TV) return;
    int m = (int)(i % TV);
    int c = (int)((i / TV) % INTER);
    int n = (int)(i / ((long)TV * INTER));
    float v = (X[i] - mean[c]) * istd[c] * g[c] + b[c];
    outBig[(((long)n * COUT + c0 + c) * Ll + l) * TV + m] = v;
}

// z[row, v] = sum_u xd[row, u] * PA[v, u]   (rows = N*INTER*T, V=25)
__global__ void adj_kernel(const float* __restrict__ xd, const float* __restrict__ PA,
                           float* __restrict__ z)
{
    long i = (long)blockIdx.x * blockDim.x + threadIdx.x;
    if (i >= (long)Nn * INTER * Tt * Vv) return;
    int v = (int)(i % Vv);
    long row = i / Vv;
    const float* xr = xd + row * Vv;
    const float* pr = PA + v * Vv;
    float s = 0.f;
    #pragma unroll
    for (int u = 0; u < Vv; ++u) s += xr[u] * pr[u];
    z[i] = s;
}

// xm[n,c,v] = mean_t xd[n,c,t,v]
__global__ void mean_t_kernel(const float* __restrict__ xd, float* __restrict__ xm)
{
    int i = blockIdx.x * blockDim.x + threadIdx.x;
    if (i >= Nn * INTER * Vv) return;
    int v = i % Vv; int nc = i / Vv;
    float s = 0.f;
    for (int t = 0; t < Tt; ++t) s += xd[((long)nc * Tt + t) * Vv + v];
    xm[i] = s * (1.f / (float)Tt);
}

// kNN by max of -||xi-xj||^2, first-index tie-break (matches lax.top_k)
__global__ __launch_bounds__(32)
void knn_kernel(const float* __restrict__ X /*[N,C,V]*/, int C, int V, int K,
                int* __restrict__ idx /*[N,V,K]*/)
{
    const int n = blockIdx.x, v0 = threadIdx.x;
    extern __shared__ float xs[];           // C*V
    for (int i = threadIdx.x; i < C * V; i += blockDim.x)
        xs[i] = X[(long)n * C * V + i];
    __syncthreads();
    if (v0 >= V) return;
    float pd[32];
    for (int u = 0; u < V; ++u) {
        float d2 = 0.f;
        for (int c = 0; c < C; ++c) {
            float d = xs[c * V + v0] - xs[c * V + u];
            d2 += d * d;
        }
        pd[u] = -d2;
    }
    for (int t = 0; t < K; ++t) {
        int best = 0; float bv = NEGINF;
        for (int u = 0; u < V; ++u) if (pd[u] > bv) { bv = pd[u]; best = u; }
        idx[((long)n * V + v0) * K + t] = best;
        pd[best] = NEGINF;
    }
}

// y[n,o,v,k] = sum_c W[o,c]*(feat-centre) + W[o,C+c]*centre
__global__ void edge_gemm_kernel(const float* __restrict__ Xm /*[N,C,V]*/,
                                 const int* __restrict__ idx /*[N,V,K]*/,
                                 const float* __restrict__ W /*[O,2C]*/,
                                 float* __restrict__ Y /*[N,O,V*K]*/,
                                 int C, int V, int K, int O, int Ntot)
{
    int i = blockIdx.x * blockDim.x + threadIdx.x;
    if (i >= Ntot) return;
    int k = i % K, v = (i / K) % V, o = (i / (K * V)) % O, n = i / (K * V * O);
    int v2 = idx[((long)n * V + v) * K + k];
    const float* xn = Xm + (long)n * C * V;
    const float* w0 = W + (long)o * 2 * C;
    float s = 0.f;
    for (int c = 0; c < C; ++c) {
        float ce = xn[c * V + v], fe = xn[c * V + v2];
        s += w0[c] * (fe - ce) + w0[C + c] * ce;
    }
    Y[i] = s;                                // i == ((n*O+o)*V+v)*K+k
}

// BN + LeakyReLU(0.2) + max over K -> Yout[N,O,V]
__global__ void edge_finalize_kernel(const float* __restrict__ Yraw /*[N,O,V*K]*/,
                                     const float* __restrict__ mean, const float* __restrict__ istd,
                                     const float* __restrict__ g, const float* __restrict__ b,
                                     float* __restrict__ Yout, int O, int V, int K, int Ntot)
{
    int i = blockIdx.x * blockDim.x + threadIdx.x;
    if (i >= Ntot) return;
    int o = (i / V) % O;
    float mx = NEGINF;
    for (int k = 0; k < K; ++k) {
        float v = (Yraw[(long)i * K + k] - mean[o]) * istd[o] * g[o] + b[o];
        v = v > 0.f ? v : 0.2f * v;
        mx = fmaxf(mx, v);
    }
    Yout[i] = mx;
}

// broadcast ye[n,c,v] over T into outBig channels [192..256), layer l
__global__ void edge_store_out_kernel(const float* __restrict__ ye, float* __restrict__ outBig, int l)
{
    int i = blockIdx.x * blockDim.x + threadIdx.x;
    if (i >= Nn * INTER * Vv) return;
    int v = i % Vv, c = (i / Vv) % INTER, n = i / (Vv * INTER);
    float val = ye[i];
    float* dst = outBig + (((long)n * COUT + 3 * INTER + c) * Ll + l) * TV + v;
    for (int t = 0; t < Tt; ++t) dst[t * Vv] = val;
}

// x_t[n,c,l,v] = max_t outBig[n,c,l,t,v]  -> [N,COUT,L*V=150]
__global__ void max_t_kernel(const float* __restrict__ outBig, float* __restrict__ x_t)
{
    int i = blockIdx.x * blockDim.x + threadIdx.x;
    if (i >= Nn * COUT * Ll * Vv) return;
    int v = i % Vv, l = (i / Vv) % Ll, nc = i / (Ll * Vv);
    const float* src = outBig + ((long)nc * Ll + l) * TV + v;
    float mx = NEGINF;
    for (int t = 0; t < Tt; ++t) mx = fmaxf(mx, src[t * Vv]);
    x_t[i] = mx;
}

// LAYERS[i] = GROUPS[i] + GROUPS[i+1], flattened
__constant__ int c_lj[44] = {0,1,2,3,20,
                             2,3,20,4,5,6,7,
                             4,5,6,7,8,9,10,11,
                             8,9,10,11,12,13,14,15,
                             12,13,14,15,16,17,18,19,
                             16,17,18,19,21,22,23,24};
__constant__ int c_lo[7] = {0,5,12,20,28,36,44};

// x_s[n,c,i] = mean over LAYERS[i] joints of relu(bn(xt_raw[n,c,i*25+joint]))
__global__ void samp_kernel(const float* __restrict__ xt_raw /*[N,C4,150]*/,
                            const float* __restrict__ mean, const float* __restrict__ istd,
                            const float* __restrict__ g, const float* __restrict__ b,
                            float* __restrict__ x_s /*[N,C4,6]*/)
{
    int i = blockIdx.x * blockDim.x + threadIdx.x;
    if (i >= Nn * C4 * Ll) return;
    int li = i % Ll, c = (i / Ll) % C4, n = i / (Ll * C4);
    int o0 = c_lo[li], o1 = c_lo[li + 1];
    const float* src = xt_raw + ((long)n * C4 + c) * (Ll * Vv) + li * Vv;
    float s = 0.f;
    for (int j = o0; j < o1; ++j) {
        float v = (src[c_lj[j]] - mean[c]) * istd[c] * g[c] + b[c];
        s += fmaxf(v, 0.f);
    }
    x_s[i] = s / (float)(o1 - o0);
}

// att -> W_agg projection + sigmoid:  sig[n,o,l]
__global__ void agg_sigmoid_kernel(const float* __restrict__ att0 /*[N,C4,6]*/,
                                   const float* __restrict__ Wagg /*[COUT,C4]*/,
                                   const float* __restrict__ bagg,
                                   float* __restrict__ sig /*[N,COUT,6]*/)
{
    int i = blockIdx.x * blockDim.x + threadIdx.x;
    if (i >= Nn * COUT * Ll) return;
    int li = i % Ll, o = (i / Ll) % COUT, n = i / (Ll * COUT);
    const float* a = att0 + (long)n * C4 * Ll + li;
    const float* w = Wagg + (long)o * C4;
    float s = bagg[o];
    for (int c = 0; c < C4; ++c) s += w[c] * a[c * Ll];
    sig[i] = 1.f / (1.f + expf(-s));
}

// osum[n,c,m] = sum_l outBig[n,c,l,m]*sig[n,c,l], fused per-channel stats.
// One block per (n,c) row; LDS reduce; one atomicAdd pair per block.
__global__ __launch_bounds__(256)
void weighted_sum_stats_kernel(const float* __restrict__ outBig,
                               const float* __restrict__ sig,
                               float* __restrict__ osum, float* __restrict__ statsBuf)
{
    const int nc = blockIdx.x;                 // 0 .. Nn*COUT-1
    const int c  = nc % COUT;
    const int tid = threadIdx.x;
    const float* src = outBig + (long)nc * Ll * TV;
    const float* sg  = sig + (long)nc * Ll;
    const float w0 = sg[0], w1 = sg[1], w2 = sg[2], w3 = sg[3], w4 = sg[4], w5 = sg[5];
    float s = 0.f, q = 0.f;
    for (int m = tid; m < TV; m += 256) {
        float v = src[m] * w0 + src[TV + m] * w1 + src[2 * TV + m] * w2
                + src[3 * TV + m] * w3 + src[4 * TV + m] * w4 + src[5 * TV + m] * w5;
        osum[(long)nc * TV + m] = v;
        s += v; q += v * v;
    }
    __shared__ float sh_s[256], sh_q[256];
    sh_s[tid] = s; sh_q[tid] = q;
    __syncthreads();
    for (int st = 128; st > 0; st >>= 1) {
        if (tid < st) { sh_s[tid] += sh_s[tid + st]; sh_q[tid] += sh_q[tid + st]; }
        __syncthreads();
    }
    if (tid == 0) {
        atomicAdd(&statsBuf[c],        sh_s[0]);
        atomicAdd(&statsBuf[COUT + c], sh_q[0]);
    }
}

// d_out = relu(bn(osum) + bn(res_raw)); res_raw lives in d_out already
__global__ void final_kernel(const float* __restrict__ osum, float* __restrict__ out,
                             const float* __restrict__ m1, const float* __restrict__ s1,
                             const float* __restrict__ g1, const float* __restrict__ b1,
                             const float* __restrict__ m2, const float* __restrict__ s2,
                             const float* __restrict__ g2, const float* __restrict__ b2)
{
    long i = (long)blockIdx.x * blockDim.x + threadIdx.x;
    if (i >= (long)Nn * COUT * TV) return;
    int o = (int)((i / TV) % COUT);
    float a = (osum[i] - m1[o]) * s1[o] * g1[o] + b1[o];
    float r = (out[i]  - m2[o]) * s2[o] * g2[o] + b2[o];
    out[i] = fmaxf(a + r, 0.f);
}

// --------------------------------------------------------------------------
static inline unsigned blk(long n, int b) { return (unsigned)((n + b - 1) / b); }

extern "C" void kernel_launch(void* const* d_in, const int* in_sizes, int n_in,
                              void* d_out, int out_size, void* d_ws, size_t ws_size,
                              hipStream_t stream)
{
    (void)in_sizes; (void)n_in; (void)out_size; (void)ws_size;
    const float* x       = (const float*)d_in[0];
    const float* PA      = (const float*)d_in[1];
    const float* W_down  = (const float*)d_in[2];
    const float* b_down  = (const float*)d_in[3];
    const float* g_down  = (const float*)d_in[4];
    const float* bt_down = (const float*)d_in[5];
    const float* W_sub   = (const float*)d_in[6];
    const float* b_sub   = (const float*)d_in[7];
    const float* g_sub   = (const float*)d_in[8];
    const float* bt_sub  = (const float*)d_in[9];
    const float* W_edge  = (const float*)d_in[10];
    const float* g_edge  = (const float*)d_in[11];
    const float* bt_edge = (const float*)d_in[12];
    const float* Wa_down = (const float*)d_in[13];
    const float* ba_down = (const float*)d_in[14];
    const float* ga_down = (const float*)d_in[15];
    const float* bta_down= (const float*)d_in[16];
    const float* Wa_edge = (const float*)d_in[17];
    const float* ga_edge = (const float*)d_in[18];
    const float* bta_edge= (const float*)d_in[19];
    const float* W_agg   = (const float*)d_in[20];
    const float* b_agg   = (const float*)d_in[21];
    const float* g_bn    = (const float*)d_in[22];
    const float* b_bn    = (const float*)d_in[23];
    const float* W_res   = (const float*)d_in[24];
    const float* b_res   = (const float*)d_in[25];
    const float* g_res   = (const float*)d_in[26];
    const float* bt_res  = (const float*)d_in[27];
    float* dout = (float*)d_out;

    // workspace bump allocator (floats)
    float* ws = (float*)d_ws;
    long off = 0;
    auto alloc = [&](long nf) { float* p = ws + off; off += nf; return p; };
    float* outBig = alloc((long)Nn * COUT * Ll * TV);   // 78.6M floats
    float* xd     = alloc((long)Nn * INTER * TV);
    float* bufA   = alloc((long)Nn * INTER * TV);
    float* bufB   = alloc((long)Nn * INTER * TV);
    float* osum   = alloc((long)Nn * COUT * TV);
    float* xm     = alloc((long)Nn * INTER * Vv);
    float* yeb    = alloc((long)Nn * INTER * Vv);
    float* yer    = alloc((long)Nn * INTER * Vv * 5);
    float* x_t    = alloc((long)Nn * COUT * Ll * Vv);
    float* xtraw  = alloc((long)Nn * C4 * Ll * Vv);
    float* x_s    = alloc((long)Nn * C4 * Ll);
    float* y2raw  = alloc((long)Nn * C4 * Ll * 3);
    float* att0   = alloc((long)Nn * C4 * Ll);
    float* sig    = alloc((long)Nn * COUT * Ll);
    float* mean1  = alloc(COUT);
    float* istd1  = alloc(COUT);
    float* mean2  = alloc(COUT);
    float* istd2  = alloc(COUT);
    float* statsB = alloc(2 * COUT);
    int*   idx1   = (int*)alloc(Nn * Vv * 5);
    int*   idx2   = (int*)alloc(Nn * Ll * 3);

    const long totI = (long)Nn * INTER * TV;    // 3.28M
    const long totO = (long)Nn * COUT * TV;     // 13.1M
    const float invTV  = 1.f / (float)(Nn * TV);        // conv stats count
    const float invAtt = 1.f / (float)(Nn * Ll * Vv);   // attention conv stats count

    for (int l = 0; l < Ll; ++l) {
        // ---- down conv (fused stats) + BN + ReLU -> xd
        hipMemsetAsync(statsB, 0, 2 * INTER * sizeof(float), stream);
        wmma_gemm_kernel<CIN, TV><<<dim3(TV / 64, INTER / 16, Nn), 32, 0, stream>>>(
            x, W_down + (long)l * INTER * CIN, b_down + l * INTER, bufB,
            INTER, (long)CIN * TV, (long)INTER * TV, statsB);
        finalize_stats_kernel<<<1, INTER, 0, stream>>>(statsB, mean1, istd1, INTER, invTV);
        bn_act_kernel<<<blk(totI, 256), 256, 0, stream>>>(
            bufB, xd, mean1, istd1, g_down + l * INTER, bt_down + l * INTER,
            INTER, TV, totI, 1);

        // ---- S adjacency branches
        for (int j = 0; j < Ss; ++j) {
            const int lj = l * Ss + j;
            adj_kernel<<<blk((long)Nn * INTER * Tt * Vv, 256), 256, 0, stream>>>(
                xd, PA + (long)lj * Vv * Vv, bufA);
            hipMemsetAsync(statsB, 0, 2 * INTER * sizeof(float), stream);
            wmma_gemm_kernel<INTER, TV><<<dim3(TV / 64, INTER / 16, Nn), 32, 0, stream>>>(
                bufA, W_sub + (long)lj * INTER * INTER, b_sub + lj * INTER, bufB,
                INTER, (long)INTER * TV, (long)INTER * TV, statsB);
            finalize_stats_kernel<<<1, INTER, 0, stream>>>(statsB, mean1, istd1, INTER, invTV);
            bn_store_out_kernel<<<blk(totI, 256), 256, 0, stream>>>(
                bufB, outBig, mean1, istd1, g_sub + lj * INTER, bt_sub + lj * INTER,
                j * INTER, l);
        }

        // ---- EdgeConv branch on T-mean
        mean_t_kernel<<<blk(Nn * INTER * Vv, 256), 256, 0, stream>>>(xd, xm);
        knn_kernel<<<Nn, 32, INTER * Vv * sizeof(float), stream>>>(xm, INTER, Vv, 5, idx1);
        edge_gemm_kernel<<<blk(Nn * INTER * Vv * 5, 256), 256, 0, stream>>>(
            xm, idx1, W_edge + (long)l * INTER * 2 * INTER, yer,
            INTER, Vv, 5, INTER, Nn * INTER * Vv * 5);
        channel_stats_kernel<<<INTER, 256, 0, stream>>>(yer, mean1, istd1, Nn, INTER, Vv * 5);
        edge_finalize_kernel<<<blk(Nn * INTER * Vv, 256), 256, 0, stream>>>(
            yer, mean1, istd1, g_edge + l * INTER, bt_edge + l * INTER,
            yeb, INTER, Vv, 5, Nn * INTER * Vv);
        edge_store_out_kernel<<<blk(Nn * INTER * Vv, 256), 256, 0, stream>>>(yeb, outBig, l);
    }

    // ---- cross-layer attention
    max_t_kernel<<<blk(Nn * COUT * Ll * Vv, 256), 256, 0, stream>>>(outBig, x_t);
    hipMemsetAsync(statsB, 0, 2 * C4 * sizeof(float), stream);
    wmma_gemm_kernel<COUT, Ll * Vv><<<dim3((Ll * Vv + 63) / 64, C4 / 16, Nn), 32, 0, stream>>>(
        x_t, Wa_down, ba_down, xtraw, C4,
        (long)COUT * Ll * Vv, (long)C4 * Ll * Vv, statsB);
    finalize_stats_kernel<<<1, C4, 0, stream>>>(statsB, mean1, istd1, C4, invAtt);
    samp_kernel<<<blk(Nn * C4 * Ll, 256), 256, 0, stream>>>(
        xtraw, mean1, istd1, ga_down, bta_down, x_s);
    knn_kernel<<<Nn, 32, C4 * Ll * sizeof(float), stream>>>(x_s, C4, Ll, 3, idx2);
    edge_gemm_kernel<<<blk(Nn * C4 * Ll * 3, 256), 256, 0, stream>>>(
        x_s, idx2, Wa_edge, y2raw, C4, Ll, 3, C4, Nn * C4 * Ll * 3);
    channel_stats_kernel<<<C4, 256, 0, stream>>>(y2raw, mean1, istd1, Nn, C4, Ll * 3);
    edge_finalize_kernel<<<blk(Nn * C4 * Ll, 256), 256, 0, stream>>>(
        y2raw, mean1, istd1, ga_edge, bta_edge, att0, C4, Ll, 3, Nn * C4 * Ll);
    agg_sigmoid_kernel<<<blk(Nn * COUT * Ll, 256), 256, 0, stream>>>(att0, W_agg, b_agg, sig);

    // ---- weighted sum (fused stats) + residual conv (fused stats) + final BN/ReLU
    hipMemsetAsync(statsB, 0, 2 * COUT * sizeof(float), stream);
    weighted_sum_stats_kernel<<<Nn * COUT, 256, 0, stream>>>(outBig, sig, osum, statsB);
    finalize_stats_kernel<<<1, COUT, 0, stream>>>(statsB, mean1, istd1, COUT, invTV);

    hipMemsetAsync(statsB, 0, 2 * COUT * sizeof(float), stream);
    wmma_gemm_kernel<CIN, TV><<<dim3(TV / 64, COUT / 16, Nn), 32, 0, stream>>>(
        x, W_res, b_res, dout, COUT, (long)CIN * TV, (long)COUT * TV, statsB);
    finalize_stats_kernel<<<1, COUT, 0, stream>>>(statsB, mean2, istd2, COUT, invTV);

    final_kernel<<<blk(totO, 256), 256, 0, stream>>>(
        osum, dout, mean1, istd1, g_bn, b_bn, mean2, istd2, g_res, bt_res);
}